// PolicyNetwork_12953621365128
// MI455X (gfx1250) — compile-verified
//
#include <hip/hip_runtime.h>
#include <cstdint>

// ---------------- problem constants ----------------
#define BATCH   32768
#define AGENTS  6
#define LIN     225
#define DENC    128
#define ADIM    64
#define NOUT    5
#define EPSG    1e-10f

// ---------------- vector types ----------------
typedef float        v2f __attribute__((ext_vector_type(2)));
typedef float        v8f __attribute__((ext_vector_type(8)));
typedef unsigned int v4u __attribute__((ext_vector_type(4)));
typedef int          v8i __attribute__((ext_vector_type(8)));
typedef int          v4i __attribute__((ext_vector_type(4)));

// ---------------- LDS geometry ----------------
// enc_w packed as K-pairs: pair-row kp holds (K=2kp, K=2kp+1) interleaved per N.
// Pair-row stride 144 pairs (288 floats): 288 % 64 banks == 32 -> the two
// half-wave B-fragment b64 reads land on disjoint bank halves.
#define ENCW_PROWS   114          // covers K=0..227 (row 113 zeroed)
#define ENCW_PSTRIDE 288
#define OBS_STRIDE   LIN          // contiguous rows == TDM write order
#define E_STRIDE     132          // 132%64==4 -> conflict-free A-frag loads
#define QK_STRIDE    68           // 68%64==4  -> conflict-free stores/reads
#define PW_FLOATS    (16*OBS_STRIDE + AGENTS*16*QK_STRIDE + AGENTS*16*8)

static __device__ __forceinline__ v8f wmma_f32x4(v2f a, v2f b, v8f c) {
  // V_WMMA_F32_16X16X4_F32 : D = A(16x4) * B(4x16) + C(16x16)
  return __builtin_amdgcn_wmma_f32_16x16x4_f32(false, a, false, b,
                                               (short)0, c, false, false);
}

__global__ __launch_bounds__(128)
void policy_fused(const float* __restrict__ obs,   // (B, A, L)
                  const float* __restrict__ enc_w, // (L, D)
                  const float* __restrict__ enc_b, // (D)
                  const float* __restrict__ q_w,   // (D, AD)
                  const float* __restrict__ k_w,   // (D, AD)
                  const float* __restrict__ fc_w,  // (2*5*D, 5)
                  const float* __restrict__ fc_b,  // (5)
                  const float* __restrict__ u,     // (B, 5)
                  float* __restrict__ out)         // (B, 1) argmax index
{
  __shared__ float s_encw[ENCW_PROWS * ENCW_PSTRIDE]; // 131,328 B
  __shared__ float s_wa[DENC * 16];                   //   8,192 B (sum of agent blocks, N-padded)
  __shared__ float s_encb[DENC];                      //     512 B
  __shared__ float s_pw[4][PW_FLOATS];                // 174,336 B (per-wave scratch)

  const int tid  = threadIdx.x;
  const int wave = tid >> 5;
  const int lane = tid & 31;
  const int half = lane >> 4;     // K-half for A/B fragments
  const int lm   = lane & 15;     // M (A) or N (B/C/D) lane index
  const int kb   = 2 * half;

  // ---------------- one-time workgroup staging ----------------
  for (int idx = tid; idx < ENCW_PROWS * DENC; idx += 128) {
    const int kp = idx >> 7, n = idx & 127;
    const int k0 = kp * 2;
    const float w0 = (k0     < LIN) ? enc_w[(size_t)k0 * DENC + n]       : 0.f;
    const float w1 = (k0 + 1 < LIN) ? enc_w[(size_t)(k0 + 1) * DENC + n] : 0.f;
    s_encw[kp * ENCW_PSTRIDE + 2 * n]     = w0;
    s_encw[kp * ENCW_PSTRIDE + 2 * n + 1] = w1;
  }
  for (int idx = tid; idx < DENC * 16; idx += 128) {
    const int k = idx >> 4, c = idx & 15;
    float s = 0.f;
    if (c < NOUT) {
      #pragma unroll
      for (int r = 0; r < 5; ++r) s += fc_w[(size_t)(r * DENC + k) * NOUT + c];
    }
    s_wa[idx] = s;
  }
  s_encb[tid] = enc_b[tid];
  __syncthreads();

  float* const pwObs = &s_pw[wave][0];                 // obs tile, later overlaid by E
  float* const pwE   = pwObs;                          // 16 x 128 (stride 132) <= obs tile
  float* const pwQK  = pwObs + 16 * OBS_STRIDE;        // [agent][16][64] stride 68
  float* const pwP   = pwQK + AGENTS * 16 * QK_STRIDE; // [agent][16][8]

  const int waveGlobal = blockIdx.x * 4 + wave;        // 1024 waves, 2048 tiles

  for (int t = 0; t < 2; ++t) {
    const int tile = waveGlobal + t * 1024;
    const int b0   = tile * 16;

    for (int ag = 0; ag < AGENTS; ++ag) {
      // -------- stage obs tile (16 rows x 225 f32) into LDS --------
      const float* gsrc = obs + ((size_t)b0 * AGENTS + ag) * LIN;
      asm volatile("s_wait_dscnt 0x0" ::: "memory");   // prior readers of this buffer done
#if __has_builtin(__builtin_amdgcn_tensor_load_to_lds)
      {
        const uint32_t ldsAddr = (uint32_t)(uintptr_t)pwObs;
        const uint64_t ga      = (uint64_t)(uintptr_t)gsrc;
        // D# group0: count=1 | lds_addr | global_addr[56:0] | type=2
        v4u g0 = { 1u, ldsAddr, (uint32_t)ga,
                   (uint32_t)((ga >> 32) & 0x01FFFFFFu) | (2u << 30) };
        // D# group1: data_size=4B, tensor 225x16, tile 225x16, dim0 stride=A*L
        v8i g1 = { (int)(2u << 16),
                   (int)((uint32_t)LIN << 16),
                   (int)(16u << 16),
                   (int)((uint32_t)LIN << 16),
                   16,
                   (int)(AGENTS * LIN),
                   0, 0 };
        v4i gz4 = { 0, 0, 0, 0 };
        v8i gz8 = { 0, 0, 0, 0, 0, 0, 0, 0 };
        // clang-23 / therock arity: (v4u, v8i, v4i, v4i, v8i, i32 cpol)
        __builtin_amdgcn_tensor_load_to_lds(g0, g1, gz4, gz4, gz8, 0);
        __builtin_amdgcn_s_wait_tensorcnt(0);
      }
#else
      for (int idx = lane; idx < 16 * LIN; idx += 32) {
        const int m = idx / LIN, c = idx - m * LIN;
        pwObs[idx] = gsrc[(size_t)m * (AGENTS * LIN) + c];
      }
#endif

      // -------- encoder GEMM: (16 x 225) @ (225 x 128), f32 WMMA --------
      v8f acc[8];
      #pragma unroll
      for (int i = 0; i < 8; ++i) acc[i] = (v8f){0.f,0.f,0.f,0.f,0.f,0.f,0.f,0.f};

      #pragma unroll 1
      for (int ks = 0; ks < 56; ++ks) {
        const int k0 = ks * 4;
        const float a0 = pwObs[lm * OBS_STRIDE + k0 + kb];
        const float a1 = pwObs[lm * OBS_STRIDE + k0 + kb + 1];
        const v2f afr = { a0, a1 };
        const int kp = (k0 >> 1) + half;
        #pragma unroll
        for (int nt = 0; nt < 8; ++nt) {
          const v2f bfr = *(const v2f*)&s_encw[kp * ENCW_PSTRIDE + 2 * (nt * 16 + lm)];
          acc[nt] = wmma_f32x4(afr, bfr, acc[nt]);
        }
      }
      { // K tail: k0 = 224, only K=224 real (rows >=225 zero)
        const int kA = 224 + kb;
        const float a0 = (kA < LIN) ? pwObs[lm * OBS_STRIDE + kA] : 0.f;
        const v2f afr = { a0, 0.f };
        const int kp = 112 + half;   // row 113 was staged as zero
        #pragma unroll
        for (int nt = 0; nt < 8; ++nt) {
          const v2f bfr = *(const v2f*)&s_encw[kp * ENCW_PSTRIDE + 2 * (nt * 16 + lm)];
          acc[nt] = wmma_f32x4(afr, bfr, acc[nt]);
        }
      }

      // -------- E = relu(acc + bias) -> LDS (overlays obs tile) --------
      #pragma unroll
      for (int nt = 0; nt < 8; ++nt) {
        const int n = nt * 16 + lm;
        const float bias = s_encb[n];
        #pragma unroll
        for (int v = 0; v < 8; ++v) {
          const int m = v + 8 * half;          // C/D layout: row = v + 8*half
          pwE[m * E_STRIDE + n] = fmaxf(acc[nt][v] + bias, 0.f);
        }
      }

      // -------- fused q/k + fc-block GEMM: (16 x 128) @ (128 x {64,5}) --------
      const float* wqk = (ag == 0) ? q_w : k_w;
      v8f aq[4], ap;
      #pragma unroll
      for (int i = 0; i < 4; ++i) aq[i] = (v8f){0.f,0.f,0.f,0.f,0.f,0.f,0.f,0.f};
      ap = (v8f){0.f,0.f,0.f,0.f,0.f,0.f,0.f,0.f};

      #pragma unroll 1
      for (int ks = 0; ks < 32; ++ks) {
        const int k0 = ks * 4;
        const int K0 = k0 + kb;
        const v2f afr = { pwE[lm * E_STRIDE + K0], pwE[lm * E_STRIDE + K0 + 1] };
        #pragma unroll
        for (int nt = 0; nt < 4; ++nt) {
          const int n = nt * 16 + lm;
          const v2f bfr = { wqk[(size_t)K0 * ADIM + n],
                            wqk[(size_t)(K0 + 1) * ADIM + n] };
          aq[nt] = wmma_f32x4(afr, bfr, aq[nt]);
        }
        v2f pfr;
        if (ag == 0) {                         // summed agent blocks (LDS, N-padded)
          pfr.x = s_wa[K0 * 16 + lm];
          pfr.y = s_wa[(K0 + 1) * 16 + lm];
        } else {                               // weighted block 5+(ag-1)
          pfr.x = (lm < NOUT) ? fc_w[(size_t)((ag + 4) * DENC + K0) * NOUT + lm] : 0.f;
          pfr.y = (lm < NOUT) ? fc_w[(size_t)((ag + 4) * DENC + K0 + 1) * NOUT + lm] : 0.f;
        }
        ap = wmma_f32x4(afr, pfr, ap);
      }

      // -------- spill q/k and P tiles to LDS --------
      float* qkDst = pwQK + ag * 16 * QK_STRIDE;
      #pragma unroll
      for (int nt = 0; nt < 4; ++nt)
        #pragma unroll
        for (int v = 0; v < 8; ++v)
          qkDst[(v + 8 * half) * QK_STRIDE + nt * 16 + lm] = aq[nt][v];
      if (lm < 8) {
        #pragma unroll
        for (int v = 0; v < 8; ++v)
          pwP[ag * 128 + (v + 8 * half) * 8 + lm] = ap[v];
      }
    } // agents

    // -------- attention softmax + gumbel argmax (1 lane per batch row) --------
    if (lane < 16) {
      const int m = lane;
      const int b = b0 + m;
      const float* qrow = pwQK + m * QK_STRIDE;
      float sc[5];
      #pragma unroll
      for (int j = 0; j < 5; ++j) {
        const float* krow = pwQK + (j + 1) * 16 * QK_STRIDE + m * QK_STRIDE;
        float s = 0.f;
        #pragma unroll 1
        for (int n = 0; n < ADIM; ++n) s += qrow[n] * krow[n];
        sc[j] = s * 0.125f;                    // 1/sqrt(64)
      }
      float mx = sc[0];
      #pragma unroll
      for (int j = 1; j < 5; ++j) mx = fmaxf(mx, sc[j]);
      float ex[5], se = 0.f;
      #pragma unroll
      for (int j = 0; j < 5; ++j) { ex[j] = __expf(sc[j] - mx); se += ex[j]; }
      const float inv = 1.f / se;

      float best = -3.4e38f; int bi = 0;
      #pragma unroll
      for (int o = 0; o < NOUT; ++o) {
        float lg = pwP[m * 8 + o];             // agent-rep part (summed blocks)
        #pragma unroll
        for (int j = 0; j < 5; ++j)
          lg += (ex[j] * inv) * pwP[(j + 1) * 128 + m * 8 + o];
        lg += fc_b[o];
        const float uu = u[(size_t)b * NOUT + o];
        const float g  = -__logf(-__logf(uu + EPSG) + EPSG);
        const float val = lg + g;              // argmax of softmax == argmax of this
        if (val > best) { best = val; bi = o; }
      }
      out[b] = (float)bi;
    }
  } // tiles
}

extern "C" void kernel_launch(void* const* d_in, const int* in_sizes, int n_in,
                              void* d_out, int out_size, void* d_ws, size_t ws_size,
                              hipStream_t stream) {
  (void)in_sizes; (void)n_in; (void)out_size; (void)d_ws; (void)ws_size;
  const float* obs   = (const float*)d_in[0];
  const float* enc_w = (const float*)d_in[1];
  const float* enc_b = (const float*)d_in[2];
  const float* q_w   = (const float*)d_in[3];
  const float* k_w   = (const float*)d_in[4];
  const float* fc_w  = (const float*)d_in[5];
  const float* fc_b  = (const float*)d_in[6];
  const float* u     = (const float*)d_in[7];
  hipLaunchKernelGGL(policy_fused, dim3(256), dim3(128), 0, stream,
                     obs, enc_w, enc_b, q_w, k_w, fc_w, fc_b, u, (float*)d_out);
}